// NodeUpdateModule_46136538693853
// MI455X (gfx1250) — compile-verified
//
#include <hip/hip_runtime.h>

// D=64, H=64, K=4, M=500000.
// Reference output: h = sum(softmax(logits), axis=1) * hvv == 1 * hvv.
// So the exact result is hvv[k,h] = sum_d hv[d] * fc_w[k,h,d], reshaped (1, K*H).
// Computed here with fp32 WMMA (V_WMMA_F32_16X16X4_F32) to keep full fp32 precision.

typedef __attribute__((ext_vector_type(2))) float v2f;
typedef __attribute__((ext_vector_type(8))) float v8f;

__global__ __launch_bounds__(512) void hvv_wmma_kernel(
    const float* __restrict__ hv,     // (1, 64)
    const float* __restrict__ fc_w,   // (4, 64, 64)  [k, h, d]
    float* __restrict__ out)          // (1, 256)     [k*64 + h]
{
    const int wave   = threadIdx.x >> 5;   // 0..15 : one wave per 16x16 output tile
    const int lane   = threadIdx.x & 31;
    const int k      = wave >> 2;          // 0..3
    const int hchunk = wave & 3;           // 0..3  (h rows hchunk*16 .. +15)
    const int mrow   = lane & 15;          // A-matrix row (h index within tile)
    const int hi     = lane >> 4;          // 0: lanes 0-15 (K0/K1), 1: lanes 16-31 (K2/K3)

    // A tile: fc_w[k][hchunk*16 + mrow][:], 16 rows x 64 cols, consumed 4 K at a time.
    const float* W = fc_w + ((size_t)k << 12) + (size_t)(hchunk * 16 + mrow) * 64;

    v8f c = {0.f, 0.f, 0.f, 0.f, 0.f, 0.f, 0.f, 0.f};

#pragma unroll
    for (int dc = 0; dc < 16; ++dc) {
        const int d = dc * 4;
        // A 16x4 fp32 layout (ISA 7.12.2): lane L(0-15): V0=K0, V1=K1 for M=L;
        //                                  lane L+16   : V0=K2, V1=K3 for M=L.
        v2f a, b;
        a.x = W[d + 2 * hi + 0];
        a.y = W[d + 2 * hi + 1];
        // B 4x16 fp32: lower lanes hold K-half {0,1} (V0=K0,V1=K1),
        //              upper lanes hold K-half {2,3} (V0=K2,V1=K3).
        // hv is broadcast across all 16 N columns, so only the K mapping matters.
        b.x = hv[d + 2 * hi + 0];
        b.y = hv[d + 2 * hi + 1];
        // D = A x B + C   (emits v_wmma_f32_16x16x4_f32)
        c = __builtin_amdgcn_wmma_f32_16x16x4_f32(
                /*neg_a=*/false, a, /*neg_b=*/false, b,
                /*c_mod=*/(short)0, c, /*reuse_a=*/false, /*reuse_b=*/false);
    }

    // D layout: VGPR r -> lanes 0-15 hold (M=r, N=lane), lanes 16-31 hold (M=r+8).
    // All N columns are identical (B was broadcast): lane 0 emits M=0..7,
    // lane 16 emits M=8..15.
    if ((lane & 15) == 0) {
        float* o = out + k * 64 + hchunk * 16 + 8 * hi;
#pragma unroll
        for (int r = 0; r < 8; ++r) o[r] = c[r];
    }
}

extern "C" void kernel_launch(void* const* d_in, const int* in_sizes, int n_in,
                              void* d_out, int out_size, void* d_ws, size_t ws_size,
                              hipStream_t stream) {
    // setup_inputs order: hv (64), msgs (500000*64, unused: softmax row-sum == 1),
    //                     fc_w (4*64*64), attn_w (4*128, unused: cancels in row-sum).
    const float* hv   = (const float*)d_in[0];
    const float* fc_w = (const float*)d_in[2];
    float* out = (float*)d_out;   // 256 floats

    hvv_wmma_kernel<<<1, 512, 0, stream>>>(hv, fc_w, out);
}